// HarebrainedPad2d_65678639891323
// MI455X (gfx1250) — compile-verified
//
#include <hip/hip_runtime.h>

typedef float v2f __attribute__((ext_vector_type(2)));
typedef float v4f __attribute__((ext_vector_type(4)));
typedef float v8f __attribute__((ext_vector_type(8)));

// ---------------------------------------------------------------------------
// Job descriptor for one edge-conv GEMM family.
// MODE 0: stride-2 down conv, 5 taps (offs -2..2), +bias, core col = x
// MODE 1: convT even cols (taps 4,2,0 @ offs -1,0,+1), core col = 2x
// MODE 2: convT odd  cols (taps 3,1   @ offs  0,+1),   core col = 2x+1
// A[co][k]: MODE0 -> wgt[co*1280 + k] (k = ci*5+t contiguous)
//           MODE1/2 -> wgt[ci*1280 + co*5 + tap[j]] (convT transpose)
// B[k][x] = inRow[ci*inChStride + ((s*x + off[j]) mod Win)], s=2 (MODE0) else 1
// ---------------------------------------------------------------------------
struct ConvJob {
  const float* inRow0;
  const float* wgt;
  const float* bias;
  float* outCore0;           // points at padded column 2 (core col 0) of first row
  int mode, Win, inChStride, inRowStride;
  int outChStride, outRowStride, Wcore, Nx;
  int nxTiles, tileStart, rsv0, rsv1;
};

struct JobTable {
  ConvJob j[12];
  int convTiles;
  int convBlocks;
};

template <int MODE, int TPC>
__device__ __forceinline__ void load_ab(const ConvJob& J, const float* __restrict__ inRow,
                                        const int* tap, const int* off,
                                        int ka, int co, int bpos, v2f& A, v2f& B) {
  const int kb = ka + 1;
  const int ci0 = ka / TPC, j0 = ka - ci0 * TPC;
  const int ci1 = kb / TPC, j1 = kb - ci1 * TPC;
  if (MODE == 0) {
    A = *(const v2f*)(J.wgt + (size_t)co * 1280 + ka);
  } else {
    A.x = J.wgt[(size_t)ci0 * 1280 + co * 5 + tap[j0]];
    A.y = J.wgt[(size_t)ci1 * 1280 + co * 5 + tap[j1]];
  }
  int p0 = bpos + off[j0];
  if (p0 < 0) p0 += J.Win;
  if (p0 >= J.Win) p0 -= J.Win;
  int p1 = bpos + off[j1];
  if (p1 < 0) p1 += J.Win;
  if (p1 >= J.Win) p1 -= J.Win;
  B.x = inRow[(size_t)ci0 * J.inChStride + p0];
  B.y = inRow[(size_t)ci1 * J.inChStride + p1];
}

template <int MODE>
__device__ __forceinline__ void conv_tile(const ConvJob& J, int xT, int coT, int row, int lane) {
  constexpr int TPC = (MODE == 0) ? 5 : (MODE == 1) ? 3 : 2;
  int tap[5] = {0, 0, 0, 0, 0};
  int off[5] = {0, 0, 0, 0, 0};
  if (MODE == 0) {
    tap[0] = 0; tap[1] = 1; tap[2] = 2; tap[3] = 3; tap[4] = 4;
    off[0] = -2; off[1] = -1; off[2] = 0; off[3] = 1; off[4] = 2;
  } else if (MODE == 1) {
    tap[0] = 4; tap[1] = 2; tap[2] = 0;
    off[0] = -1; off[1] = 0; off[2] = 1;
  } else {
    tap[0] = 3; tap[1] = 1;
    off[0] = 0; off[1] = 1;
  }

  const int h = lane >> 4;   // ISA 16x16x4 layout: lane(h,l): A[l][2h+v], B[2h+v][l]
  const int l = lane & 15;
  const float* __restrict__ inRow = J.inRow0 + (size_t)row * J.inRowStride;
  float* __restrict__ outRow = J.outCore0 + (size_t)row * J.outRowStride;
  const int x = xT * 16 + l;
  const int co = coT * 16 + l;
  const int bpos = (MODE == 0) ? (2 * x) : x;

  v8f acc0 = {0.f, 0.f, 0.f, 0.f, 0.f, 0.f, 0.f, 0.f};
  v8f acc1 = acc0;
  const int Kh = 128 * TPC;  // half of Ktot = 256*TPC; both halves divisible by 4
  for (int k0 = 0; k0 < Kh; k0 += 4) {
    v2f A0, B0, A1, B1;
    load_ab<MODE, TPC>(J, inRow, tap, off, k0 + 2 * h, co, bpos, A0, B0);
    load_ab<MODE, TPC>(J, inRow, tap, off, k0 + Kh + 2 * h, co, bpos, A1, B1);
    acc0 = __builtin_amdgcn_wmma_f32_16x16x4_f32(false, A0, false, B0, (short)0, acc0, false, false);
    acc1 = __builtin_amdgcn_wmma_f32_16x16x4_f32(false, A1, false, B1, (short)0, acc1, false, false);
  }
  v8f acc = acc0 + acc1;

  // Store D[M=j+8h][N=l]; outCore0 points at padded column 2 (core col 0).
  if (x < J.Nx) {
    const int c = (MODE == 0) ? x : (2 * x + ((MODE == 2) ? 1 : 0));
#pragma unroll
    for (int jj = 0; jj < 8; ++jj) {
      const int m = coT * 16 + jj + 8 * h;
      float v = acc[jj];
      if (MODE == 0) v += J.bias[m];
      float* bp = outRow + (size_t)m * J.outChStride;
      bp[c] = v;
      if (c < 2) bp[c + J.Wcore] = v;             // right circular pad cols W+2,W+3
      if (c >= J.Wcore - 2) bp[c - J.Wcore] = v;  // left  circular pad cols 0,1
    }
  }
}

// Strip geometry (fixed by the problem)
#define NV0 792064    // 256*34*364/4
#define NV1 2965504   // 256*64*724/4
#define NV2 17004544  // 256*184*1444/4
#define PB0 (NV0)
#define PB1 (PB0 + NV1)
#define PB2 (PB1 + NV2)
#define PB3 (PB2 + NV1)
#define PB4 (PB3 + NV0)  // 24519680 float4s total

__global__ __launch_bounds__(256) void fused_kernel(
    JobTable jt,
    const float* __restrict__ s0, const float* __restrict__ s1, const float* __restrict__ s2,
    const float* __restrict__ s3, const float* __restrict__ s4, float* __restrict__ out) {
  if ((int)blockIdx.x < jt.convBlocks) {
    // ---------------- conv tiles: one wave per 16x16 output tile ----------------
    const int wave = threadIdx.x >> 5;
    const int lane = threadIdx.x & 31;
    const int tile = blockIdx.x * 8 + wave;
    if (tile >= jt.convTiles) return;  // wave-uniform
    // wave-uniform job select (unrolled conditional copy -> s_cselect, no scratch)
    ConvJob J = jt.j[0];
#pragma unroll
    for (int i = 1; i < 12; ++i)
      if (tile >= jt.j[i].tileStart) J = jt.j[i];
    const int lt = tile - J.tileStart;
    const int xT = lt % J.nxTiles;
    const int rest = lt / J.nxTiles;
    const int coT = rest & 15;
    const int row = rest >> 4;
    if (J.mode == 0) conv_tile<0>(J, xT, coT, row, lane);
    else if (J.mode == 1) conv_tile<1>(J, xT, coT, row, lane);
    else conv_tile<2>(J, xT, coT, row, lane);
  } else {
    // ---------------- bulk circular-pad copy, float4 per thread ----------------
    const int vec = (blockIdx.x - jt.convBlocks) * 256 + threadIdx.x;
    if (vec >= PB4) return;
    int st, start;
    if (vec < PB0)      { st = 0; start = 0;   }
    else if (vec < PB1) { st = 1; start = PB0; }
    else if (vec < PB2) { st = 2; start = PB1; }
    else if (vec < PB3) { st = 3; start = PB2; }
    else                { st = 4; start = PB3; }
    const int H = (st == 2) ? 180 : ((st == 1 || st == 3) ? 60 : 30);
    const int W = (st == 2) ? 1440 : ((st == 1 || st == 3) ? 720 : 360);
    const int Hp = H + 4, wq = (W + 4) >> 2;
    const int lv = vec - start;
    const int jq = lv % wq;
    const int t = lv / wq;
    const int r = t % Hp;
    const int c = t / Hp;
    v4f ov = {0.f, 0.f, 0.f, 0.f};
    if (r >= 2 && r < Hp - 2) {
      const float* sp = (st == 0) ? s0 : (st == 1) ? s1 : (st == 2) ? s2 : (st == 3) ? s3 : s4;
      const float* src = sp + ((size_t)c * H + (r - 2)) * W;
      const int j4 = jq * 4;
#pragma unroll
      for (int e = 0; e < 4; ++e) {
        int jj = j4 + e - 2;
        if (jj < 0) jj += W;
        if (jj >= W) jj -= W;
        ov[e] = src[jj];
      }
    } else if (!((st == 0 && r < 2) || (st == 4 && r >= Hp - 2))) {
      return;  // edge rows owned (fully overwritten) by conv tiles in this same launch
    }
    *(v4f*)(out + (size_t)vec * 4) = ov;
  }
}

extern "C" void kernel_launch(void* const* d_in, const int* in_sizes, int n_in,
                              void* d_out, int out_size, void* d_ws, size_t ws_size,
                              hipStream_t stream) {
  (void)in_sizes; (void)n_in; (void)out_size; (void)d_ws; (void)ws_size;
  const float* s[5];
  for (int i = 0; i < 5; ++i) s[i] = (const float*)d_in[i];
  const float* wi0 = (const float*)d_in[5];
  const float* wi1 = (const float*)d_in[6];
  const float* wo0 = (const float*)d_in[7];
  const float* bo0 = (const float*)d_in[8];
  const float* wo1 = (const float*)d_in[9];
  const float* bo1 = (const float*)d_in[10];
  float* out = (float*)d_out;

  const int C = 256;
  const int Hs[5] = {30, 60, 180, 60, 30};
  const int Ws[5] = {360, 720, 1440, 720, 360};
  size_t off[5];
  size_t acc = 0;
  for (int i = 0; i < 5; ++i) { off[i] = acc; acc += (size_t)C * (Hs[i] + 4) * (Ws[i] + 4); }

  auto chS = [&](int i) { return (Hs[i] + 4) * (Ws[i] + 4); };
  auto inS = [&](int i) { return Hs[i] * Ws[i]; };

  JobTable jt;
  int ts = 0;
  auto mkjob = [&](int slot, int mode, const float* inRow0, int inCh, int inRS, int Win,
                   const float* w, const float* b, float* core0, int outCh, int outRS,
                   int Wcore, int Nx) {
    ConvJob& J = jt.j[slot];
    J.inRow0 = inRow0; J.wgt = w; J.bias = b; J.outCore0 = core0;
    J.mode = mode; J.Win = Win; J.inChStride = inCh; J.inRowStride = inRS;
    J.outChStride = outCh; J.outRowStride = outRS; J.Wcore = Wcore; J.Nx = Nx;
    J.nxTiles = (Nx + 15) / 16; J.tileStart = ts; J.rsv0 = 0; J.rsv1 = 0;
    ts += J.nxTiles * 16 * 2;  // x-tiles * 16 co-tiles * 2 rows
  };

  // down convs (mode 0)
  mkjob(0, 0, s[2], inS(2), 1440, 1440, wo0, bo0,
        out + off[1] + (size_t)62 * 724 + 2, chS(1), 724, 720, 720);   // s2 top -> sp1 rows 62..63
  mkjob(1, 0, s[2] + (size_t)178 * 1440, inS(2), 1440, 1440, wo0, bo0,
        out + off[3] + 2, chS(3), 724, 720, 720);                      // s2 bot -> sp3 rows 0..1
  mkjob(2, 0, s[1], inS(1), 720, 720, wo1, bo1,
        out + off[0] + (size_t)32 * 364 + 2, chS(0), 364, 360, 360);   // s1 top -> sp0 rows 32..33
  mkjob(3, 0, s[3] + (size_t)58 * 720, inS(3), 720, 720, wo1, bo1,
        out + off[4] + 2, chS(4), 364, 360, 360);                      // s3 bot -> sp4 rows 0..1
  // up transposed convs (mode 1 = even cols, mode 2 = odd cols)
  mkjob(4, 1, s[1] + (size_t)58 * 720, inS(1), 720, 720, wi0, nullptr,
        out + off[2] + 2, chS(2), 1444, 1440, 720);                    // s1 bot -> sp2 rows 0..1
  mkjob(5, 2, s[1] + (size_t)58 * 720, inS(1), 720, 720, wi0, nullptr,
        out + off[2] + 2, chS(2), 1444, 1440, 720);
  mkjob(6, 1, s[3], inS(3), 720, 720, wi0, nullptr,
        out + off[2] + (size_t)182 * 1444 + 2, chS(2), 1444, 1440, 720);  // s3 top -> sp2 rows 182..183
  mkjob(7, 2, s[3], inS(3), 720, 720, wi0, nullptr,
        out + off[2] + (size_t)182 * 1444 + 2, chS(2), 1444, 1440, 720);
  mkjob(8, 1, s[0] + (size_t)28 * 360, inS(0), 360, 360, wi1, nullptr,
        out + off[1] + 2, chS(1), 724, 720, 360);                      // s0 bot -> sp1 rows 0..1
  mkjob(9, 2, s[0] + (size_t)28 * 360, inS(0), 360, 360, wi1, nullptr,
        out + off[1] + 2, chS(1), 724, 720, 360);
  mkjob(10, 1, s[4], inS(4), 360, 360, wi1, nullptr,
        out + off[3] + (size_t)62 * 724 + 2, chS(3), 724, 720, 360);   // s4 top -> sp3 rows 62..63
  mkjob(11, 2, s[4], inS(4), 360, 360, wi1, nullptr,
        out + off[3] + (size_t)62 * 724 + 2, chS(3), 724, 720, 360);

  jt.convTiles = ts;                       // 13056
  jt.convBlocks = (ts + 7) / 8;            // 8 waves (tiles) per 256-thread block

  const int padVec = PB4;                  // 24,519,680 float4s
  const int padBlocks = (padVec + 255) / 256;
  dim3 grid(jt.convBlocks + padBlocks), blk(256);
  fused_kernel<<<grid, blk, 0, stream>>>(jt, s[0], s[1], s[2], s[3], s[4], out);
}